// HBSLinear_35845797052977
// MI455X (gfx1250) — compile-verified
//
#include <hip/hip_runtime.h>
#include <cstdint>
#include <cstddef>

// ---------------- problem constants ----------------
#define N_COLS   4096
#define N_BATCH  16384
#define N_STAGES 12
#define N_PAIRS  2048          // N_COLS / 2
#define ROWS     16            // batch rows per workgroup (= WMMA M)
#define RS       4100          // padded LDS row stride in floats (4100 % 64 == 4 -> conflict-free column access)
#define TPB      256           // 8 waves of 32
#define TRIG_N   (N_STAGES * N_PAIRS)

typedef float v2f __attribute__((ext_vector_type(2)));
typedef float v8f __attribute__((ext_vector_type(8)));

#if defined(__gfx1250__) && __has_builtin(__builtin_amdgcn_wmma_f32_16x16x4_f32)
#define HAVE_WMMA_F32 1
#else
#define HAVE_WMMA_F32 0
#endif

// ---------------- CDNA5 async LDS<->global helpers (CDNA5 ISA) ----------------
__device__ __forceinline__ void async_load_b128(unsigned lds_off, const void* gaddr) {
#if defined(__gfx1250__)
  asm volatile("global_load_async_to_lds_b128 %0, %1, off"
               :: "v"(lds_off), "v"(gaddr) : "memory");
#endif
}
__device__ __forceinline__ void async_store_b128(void* gaddr, unsigned lds_off) {
#if defined(__gfx1250__)
  asm volatile("global_store_async_from_lds_b128 %0, %1, off"
               :: "v"(gaddr), "v"(lds_off) : "memory");
#endif
}
__device__ __forceinline__ void wait_async0() {
#if defined(__gfx1250__)
  asm volatile("s_wait_asynccnt 0" ::: "memory");
#endif
}

// ---------------- precompute 1: cos/sin tables (12 x 2048) ----------------
__global__ void hbs_trig_kernel(const float* __restrict__ ang,
                                float* __restrict__ ctab,
                                float* __restrict__ stab) {
  int i = blockIdx.x * blockDim.x + threadIdx.x;
  if (i < TRIG_N) {
    float th = ang[i];
    ctab[i] = cosf(th);
    stab[i] = sinf(th);
  }
}

// ---------------- precompute 2: combined stage 0-3 block matrices ----------------
// Stages 0..3 (dist 1,2,4,8) act entirely inside aligned 16-column blocks.
// Build W_b = S3*S2*S1*S0 (16x16 per block b, 256 blocks), store transposed:
// wt[b*256 + k*16 + n] = W_b[n][k]  (i.e. W^T rows indexed by source column k).
__global__ void hbs_wmat_kernel(const float* __restrict__ ang,
                                float* __restrict__ wt) {
  const int b = threadIdx.x;       // 256 threads, one block matrix each
  float W[16][16];
  for (int i = 0; i < 16; ++i)
    for (int j = 0; j < 16; ++j)
      W[i][j] = (i == j) ? 1.0f : 0.0f;

  for (int s = 0; s < 4; ++s) {
    const int dist = 1 << s;
    for (int g = 0; g < 16 / (2 * dist); ++g) {
      for (int j = 0; j < dist; ++j) {
        const int li0 = g * 2 * dist + j;
        const int li1 = li0 + dist;
        const int gi0 = b * 16 + li0;
        const int p   = ((gi0 >> (s + 1)) << s) | (gi0 & (dist - 1));
        const float th = ang[s * N_PAIRS + p];
        const float c  = cosf(th);
        const float si = sinf(th);
        for (int k = 0; k < 16; ++k) {
          const float a0 = W[li0][k], a1 = W[li1][k];
          W[li0][k] = c * a0 - si * a1;
          W[li1][k] = si * a0 + c * a1;
        }
      }
    }
  }
  for (int k = 0; k < 16; ++k)
    for (int n = 0; n < 16; ++n)
      wt[b * 256 + k * 16 + n] = W[n][k];
}

// ---------------- main fused butterfly kernel ----------------
// One workgroup owns a 16-row x 4096-col tile resident in LDS for all 12 stages.
extern "C" __global__ void __launch_bounds__(TPB, 1)
hbs_butterfly_kernel(const float* __restrict__ x,
                     const float* __restrict__ ctab,
                     const float* __restrict__ stab,
                     const float* __restrict__ wt,
                     float* __restrict__ out) {
  extern __shared__ float tile[];            // ROWS * RS floats (~256.25 KB)
  const int tid  = threadIdx.x;
  const int row0 = blockIdx.x * ROWS;

  // ---- copy-in: 16 rows via async global->LDS b128 (4096 float4 chunks) ----
  for (int k = 0; k < 16; ++k) {
    const int lin = tid + (k << 8);          // 0..4095
    const int r   = lin >> 8;                // row in tile
    const int ch  = lin & 255;               // float4 chunk in row
    const float* g = x + (size_t)(row0 + r) * N_COLS + ch * 4;
    const unsigned lo = (unsigned)(size_t)&tile[r * RS + ch * 4];
    async_load_b128(lo, (const void*)g);
  }
  wait_async0();
  __syncthreads();

#if HAVE_WMMA_F32
  // ---- stages 0..3 fused as block-diagonal 16x16 GEMM via V_WMMA_F32_16X16X4_F32 ----
  {
    const int wave = tid >> 5;
    const int lane = tid & 31;
    const int m    = lane & 15;              // A: row M / B: col N / D: col N
    const int hi   = lane >> 4;              // upper half-wave handles K+2, M+8
    for (int t = 0; t < 32; ++t) {           // 8 waves x 32 blocks = 256 blocks
      const int b = wave * 32 + t;
      const float* wtb = wt + b * 256;
      v8f acc = {0.f, 0.f, 0.f, 0.f, 0.f, 0.f, 0.f, 0.f};
#pragma unroll
      for (int kc = 0; kc < 16; kc += 4) {   // K = 16 as four 16x16x4 WMMAs
        const int kk = kc + 2 * hi;
        v2f av, bv;
        av[0] = tile[m * RS + b * 16 + kk];      // A[M=m][K=kk]
        av[1] = tile[m * RS + b * 16 + kk + 1];  // A[M=m][K=kk+1]
        bv[0] = wtb[kk * 16 + m];                // B=W^T [K=kk][N=m]
        bv[1] = wtb[(kk + 1) * 16 + m];
        acc = __builtin_amdgcn_wmma_f32_16x16x4_f32(
            false, av, false, bv, (short)0, acc, false, false);
      }
#pragma unroll
      for (int i = 0; i < 8; ++i)              // D vgpr i: M = i + 8*hi, N = m
        tile[(i + 8 * hi) * RS + b * 16 + m] = acc[i];
    }
  }
  __syncthreads();
  const int s_start = 4;
#else
  const int s_start = 0;                       // fallback: all stages as LDS butterflies
#endif

  // ---- remaining stages: LDS butterflies, cos/sin reused across all 16 rows ----
  for (int s = s_start; s < N_STAGES; ++s) {
    const int dist = 1 << s;
    for (int k = 0; k < N_PAIRS / TPB; ++k) {  // 8 pairs / thread / stage
      const int p  = tid + (k << 8);
      const int j  = p & (dist - 1);
      const int i0 = ((p >> s) << (s + 1)) | j;
      const int i1 = i0 + dist;
      const float c  = ctab[s * N_PAIRS + p];
      const float si = stab[s * N_PAIRS + p];
#pragma unroll 4
      for (int r = 0; r < ROWS; ++r) {
        float* p0 = &tile[r * RS + i0];
        float* p1 = &tile[r * RS + i1];
        const float v0 = *p0, v1 = *p1;
        *p0 = fmaf(c,  v0, -si * v1);
        *p1 = fmaf(si, v0,  c  * v1);
      }
    }
    __syncthreads();
  }

  // ---- copy-out: async LDS->global b128 ----
  for (int k = 0; k < 16; ++k) {
    const int lin = tid + (k << 8);
    const int r   = lin >> 8;
    const int ch  = lin & 255;
    float* g = out + (size_t)(row0 + r) * N_COLS + ch * 4;
    const unsigned lo = (unsigned)(size_t)&tile[r * RS + ch * 4];
    async_store_b128((void*)g, lo);
  }
  wait_async0();
}

// ---------------- launch ----------------
extern "C" void kernel_launch(void* const* d_in, const int* in_sizes, int n_in,
                              void* d_out, int out_size, void* d_ws, size_t ws_size,
                              hipStream_t stream) {
  const float* x   = (const float*)d_in[0];       // (16384, 4096) f32
  const float* ang = (const float*)d_in[1];       // (12, 2048) f32
  float* out = (float*)d_out;

  // workspace layout: ctab | stab | wt  (24576 + 24576 + 65536 floats = 448 KB)
  float* ctab = (float*)d_ws;
  float* stab = ctab + TRIG_N;
  float* wt   = stab + TRIG_N;

  hbs_trig_kernel<<<(TRIG_N + TPB - 1) / TPB, TPB, 0, stream>>>(ang, ctab, stab);
  hbs_wmat_kernel<<<1, 256, 0, stream>>>(ang, wt);

  const size_t lds_bytes = (size_t)ROWS * RS * sizeof(float);   // 262400 B < 320 KB
  hbs_butterfly_kernel<<<dim3(N_BATCH / ROWS), dim3(TPB), lds_bytes, stream>>>(
      x, ctab, stab, wt, out);
}